// TreeEnergyLoss2_66151086293303
// MI455X (gfx1250) — compile-verified
//
#include <hip/hip_runtime.h>
#include <hip/hip_bf16.h>

typedef __attribute__((ext_vector_type(2))) float v2f;
typedef __attribute__((ext_vector_type(8))) float v8f;

#define NPIX 4096          // 64*64
#define NEDGE 8064         // 63*64 + 64*63
#define NSORT 8192
#define BATCH 8
#define NCLS 21

// ---------------------------------------------------------------------------
// Resize as GEMM:  out(64x64) = Rh(64x256) * X(256x256) * RwT(256x64), f32 WMMA
// ---------------------------------------------------------------------------
__global__ void tel_zero_kernel(float* p, long n) {
    long i = (long)blockIdx.x * 256 + threadIdx.x;
    if (i < n) p[i] = 0.f;
}

__global__ void tel_interp_kernel(float* Rh, float* RwT) {
    int r = threadIdx.x;
    if (r < 64) {
        // half-pixel (align_corners=False): src = (r+0.5)*4 - 0.5
        float f = (r + 0.5f) * 4.0f - 0.5f;
        int s0 = (int)floorf(f);
        float t = f - (float)s0;
        if (s0 < 0) { s0 = 0; t = 0.f; }
        int s1 = s0 + 1; if (s1 > 255) s1 = 255;
        Rh[r * 256 + s0] += 1.0f - t;
        Rh[r * 256 + s1] += t;
        RwT[s0 * 64 + r] += 1.0f - t;
        RwT[s1 * 64 + r] += t;
    }
}

// One wave computes one 16x16 tile; K stepped by 4 via v_wmma_f32_16x16x4_f32.
__global__ void __launch_bounds__(32) tel_gemm_wmma_f32(
    const float* __restrict__ A, const float* __restrict__ B, float* __restrict__ C,
    int N, int K, int lda, int ldb, int ldc, long sA, long sB, long sC) {
    const float* Ab = A + (long)blockIdx.y * sA;
    const float* Bb = B + (long)blockIdx.y * sB;
    float* Cb = C + (long)blockIdx.y * sC;
    int tilesN = N >> 4;
    int tm = (blockIdx.x / tilesN) << 4;
    int tn = (blockIdx.x % tilesN) << 4;
    int lane = threadIdx.x;
    int half = lane >> 4;      // which half of the wave
    int l = lane & 15;
    const float* arow = Ab + (long)(tm + l) * lda;   // A: M across lanes 0..15
    v8f acc = {};
    for (int k0 = 0; k0 < K; k0 += 4) {
        int ka = k0 + 2 * half;                      // lanes 16..31 take K=2,3
        __builtin_prefetch(arow + k0 + 16, 0, 1);    // global_prefetch_b8
        v2f a, b;
        a.x = arow[ka];
        a.y = arow[ka + 1];
        b.x = Bb[(long)ka * ldb + tn + l];           // B: N across lanes
        b.y = Bb[(long)(ka + 1) * ldb + tn + l];
        acc = __builtin_amdgcn_wmma_f32_16x16x4_f32(
            false, a, false, b, (short)0, acc, false, false);
    }
#pragma unroll
    for (int g = 0; g < 8; ++g)
        Cb[(long)(tm + 8 * half + g) * ldc + tn + l] = acc[g];
}

// ---------------------------------------------------------------------------
// ROI nearest downsample: src = (dst*256)//64 = 4*dst
// ---------------------------------------------------------------------------
__global__ void tel_roi_kernel(const int* __restrict__ rois, float* __restrict__ roi_ds) {
    int gi = blockIdx.x * 256 + threadIdx.x;
    if (gi >= BATCH * NPIX) return;
    int b = gi >> 12, p = gi & 4095;
    int r = p >> 6, c = p & 63;
    roi_ds[gi] = (float)rois[(long)b * 65536 + (r * 4) * 256 + c * 4];
}

// ---------------------------------------------------------------------------
// MST build: one workgroup per image. Edge weights -> LDS bitonic sort of
// stable 64-bit keys -> serial Kruskal (union-find) -> serial BFS.
// Dynamic LDS = 180224 bytes (needs CDNA5's 320KB/WGP LDS).
// ---------------------------------------------------------------------------
__global__ void __launch_bounds__(256) tel_build_tree(
    const float* __restrict__ embed, int nch, long bstride,
    int* __restrict__ order_g, int* __restrict__ par_g) {
    extern __shared__ char smem[];
    unsigned long long* key = (unsigned long long*)smem;        // [8192]  64KB
    int* uf   = (int*)(smem + 65536);                           // [4096]
    int* head = (int*)(smem + 81920);                           // [4096]
    int* tail = (int*)(smem + 98304);                           // [4096]
    int* enxt = (int*)(smem + 114688);                          // [8192]
    int* eto  = (int*)(smem + 147456);                          // [8192]
    int b = blockIdx.x;
    const float* e = embed + (long)b * bstride;
    int tid = threadIdx.x;

    for (int i = tid; i < NPIX; i += 256) { uf[i] = i; head[i] = -1; tail[i] = -1; }

    // edge weights (double accumulation like the f64 reference), stable keys
    for (int s = tid; s < NSORT; s += 256) {
        unsigned long long kk;
        if (s < NEDGE) {
            int u, v;
            if (s < 4032) { u = s; v = s + 64; }                       // vertical
            else { int e2 = s - 4032; int y = e2 / 63; int x = e2 - y * 63;
                   u = y * 64 + x; v = u + 1; }                        // horizontal
            double d = 0.0;
            for (int c = 0; c < nch; ++c) {
                double du = (double)e[(long)c * NPIX + u] - (double)e[(long)c * NPIX + v];
                d += du * du;
            }
            unsigned wb = __float_as_uint((float)d);                   // >=0: order-preserving
            kk = ((unsigned long long)wb << 32) | (unsigned)s;         // stable tie-break
        } else kk = ~0ull;
        key[s] = kk;
    }
    __syncthreads();

    // bitonic sort ascending, 8192 u64 keys
    for (unsigned k = 2; k <= NSORT; k <<= 1) {
        for (unsigned j = k >> 1; j > 0; j >>= 1) {
            for (unsigned i = tid; i < NSORT; i += 256) {
                unsigned ij = i ^ j;
                if (ij > i) {
                    unsigned long long a = key[i], c = key[ij];
                    bool asc = ((i & k) == 0);
                    if ((a > c) == asc) { key[i] = c; key[ij] = a; }
                }
            }
            __syncthreads();
        }
    }

    // serial Kruskal, adjacency as append-ordered linked lists
    if (tid == 0) {
        int cnt = 0;
        for (int s = 0; s < NSORT && cnt < NPIX - 1; ++s) {
            int idx = (int)(key[s] & 0xffffffffu);
            if (idx >= NEDGE) continue;
            int u, v;
            if (idx < 4032) { u = idx; v = idx + 64; }
            else { int e2 = idx - 4032; int y = e2 / 63; int x = e2 - y * 63;
                   u = y * 64 + x; v = u + 1; }
            int ra = u; while (uf[ra] != ra) { uf[ra] = uf[uf[ra]]; ra = uf[ra]; }
            int rb = v; while (uf[rb] != rb) { uf[rb] = uf[uf[rb]]; rb = uf[rb]; }
            if (ra != rb) {
                uf[ra] = rb;
                int s0 = 2 * cnt, s1 = 2 * cnt + 1;
                eto[s0] = v; enxt[s0] = -1;
                if (head[u] < 0) head[u] = s0; else enxt[tail[u]] = s0;
                tail[u] = s0;
                eto[s1] = u; enxt[s1] = -1;
                if (head[v] < 0) head[v] = s1; else enxt[tail[v]] = s1;
                tail[v] = s1;
                ++cnt;
            }
        }
    }
    __syncthreads();
    for (int i = tid; i < NPIX; i += 256) uf[i] = 0;   // reuse as 'seen'
    __syncthreads();

    // serial BFS from node 0 (queue reuses key LDS region)
    if (tid == 0) {
        int* queue = (int*)key;
        int* seen = uf;
        int hq = 0, cq = 1;
        queue[0] = 0; seen[0] = 1;
        order_g[(long)b * NPIX] = 0;
        par_g[(long)b * NPIX] = 0;
        while (hq < cq) {
            int x = queue[hq++];
            for (int s = head[x]; s >= 0; s = enxt[s]) {
                int y = eto[s];
                if (!seen[y]) {
                    seen[y] = 1;
                    par_g[(long)b * NPIX + y] = x;
                    order_g[(long)b * NPIX + cq] = y;
                    queue[cq++] = y;
                }
            }
        }
    }
}

// ---------------------------------------------------------------------------
// Edge weights for the filter: w[i] = exp(-||e_i - e_par(i)||^2 / sigma)
// ---------------------------------------------------------------------------
__global__ void tel_wedge_kernel(const float* __restrict__ embed, int nch, long bstride,
                                 const int* __restrict__ par_g, float sigma,
                                 float* __restrict__ wout) {
    int gi = blockIdx.x * 256 + threadIdx.x;
    if (gi >= BATCH * NPIX) return;
    int b = gi >> 12, i = gi & 4095;
    const float* e = embed + (long)b * bstride;
    int p = par_g[gi];
    float d = 0.f;
    for (int c = 0; c < nch; ++c) {
        float du = e[(long)c * NPIX + i] - e[(long)c * NPIX + p];
        d += du * du;
    }
    wout[gi] = expf(-d / sigma);
}

// ---------------------------------------------------------------------------
// Tree filter: one block per batch image, 11 channels per chunk across lanes.
// aggr[4096][11] lives in LDS (180KB). Up-sweep then in-place down-sweep run
// lockstep inside wave 0 (LDS ops are in-order per wave).
// Dynamic LDS = 180224 + 3*16384 = 229376 bytes.
// ---------------------------------------------------------------------------
__global__ void __launch_bounds__(256) tel_tree_filter_chunk(
    const float* __restrict__ fin, int mode, int ch0,
    const int* __restrict__ order_g, const int* __restrict__ par_g,
    const float* __restrict__ wedge_g, float* __restrict__ gtmp) {
    extern __shared__ char smem[];
    float* aggr = (float*)smem;                         // [4096*11]
    int*   sord = (int*)(smem + 180224);                // [4096]
    int*   spar = (int*)(smem + 196608);                // [4096]
    float* swe  = (float*)(smem + 212992);              // [4096]
    int b = blockIdx.x, tid = threadIdx.x;
    const int* ob = order_g + (long)b * NPIX;
    const int* pb = par_g + (long)b * NPIX;
    const float* wb = wedge_g + (long)b * NPIX;

    for (int i = tid; i < NPIX; i += 256) {
        sord[i] = ob[i]; spar[i] = pb[i]; swe[i] = wb[i];
        for (int cc = 0; cc < 11; ++cc) {
            int ch = ch0 + cc;
            float val;
            if (ch >= NCLS) val = 1.0f;                               // norm channel
            else if (mode == 0) val = fin[((long)b * NCLS + ch) * NPIX + i];   // [B,C,N]
            else val = fin[((long)b * NPIX + i) * NCLS + ch];                  // [B,N,C]
            aggr[i * 11 + cc] = val;
        }
    }
    __syncthreads();

    if (tid < 11) {
        // up-sweep: leaf -> root over reversed BFS order
        for (int s = NPIX - 1; s >= 1; --s) {
            int v = sord[s]; int p = spar[v]; float w = swe[v];
            aggr[p * 11 + tid] += w * aggr[v * 11 + tid];
        }
        // down-sweep (in place; parent precedes child in BFS order)
        for (int s = 1; s < NPIX; ++s) {
            int v = sord[s]; int p = spar[v]; float w = swe[v];
            float av = aggr[v * 11 + tid];
            aggr[v * 11 + tid] = av + w * (aggr[p * 11 + tid] - w * av);
        }
    }
    __syncthreads();

    for (int i = tid; i < NPIX; i += 256)
        for (int cc = 0; cc < 11; ++cc)
            gtmp[((long)b * NPIX + i) * 22 + ch0 + cc] = aggr[i * 11 + cc];
}

__global__ void tel_divide_kernel(const float* __restrict__ gtmp, float* __restrict__ AS) {
    long idx = (long)blockIdx.x * 256 + threadIdx.x;   // over B*N*21
    if (idx >= (long)BATCH * NPIX * NCLS) return;
    long pix = idx / NCLS; int c = (int)(idx - pix * NCLS);
    AS[idx] = gtmp[pix * 22 + c] / gtmp[pix * 22 + NCLS];
}

// ---------------------------------------------------------------------------
// Deterministic two-stage loss reduction
// ---------------------------------------------------------------------------
__global__ void __launch_bounds__(256) tel_loss_partial(
    const float* __restrict__ pred_pred, const float* __restrict__ AS,
    const float* __restrict__ roi, float* __restrict__ partials) {
    __shared__ float sd[256];
    const long total = (long)BATCH * NCLS * NPIX;
    float acc = 0.f;
    for (long idx = (long)blockIdx.x * 256 + threadIdx.x; idx < total;
         idx += (long)gridDim.x * 256) {
        long b = idx / ((long)NCLS * NPIX);
        long r = idx - b * (long)NCLS * NPIX;
        int c = (int)(r / NPIX);
        int i = (int)(r - (long)c * NPIX);
        float rv = roi[b * NPIX + i];
        float as = AS[(b * NPIX + i) * NCLS + c];
        acc += rv * fabsf(pred_pred[idx] - as);
    }
    sd[threadIdx.x] = acc; __syncthreads();
    for (int s = 128; s > 0; s >>= 1) {
        if (threadIdx.x < s) sd[threadIdx.x] += sd[threadIdx.x + s];
        __syncthreads();
    }
    if (threadIdx.x == 0) partials[blockIdx.x] = sd[0];
}

__global__ void __launch_bounds__(256) tel_loss_final(
    const float* __restrict__ partials, const float* __restrict__ roi,
    const float* __restrict__ tw, float* __restrict__ out) {
    __shared__ float sd[256];
    float a = 0.f;
    for (int i = threadIdx.x; i < 512; i += 256) a += partials[i];
    float nvp = 0.f;
    for (int i = threadIdx.x; i < BATCH * NPIX; i += 256) nvp += roi[i];
    sd[threadIdx.x] = a; __syncthreads();
    for (int s = 128; s > 0; s >>= 1) {
        if (threadIdx.x < s) sd[threadIdx.x] += sd[threadIdx.x + s];
        __syncthreads();
    }
    float loss = sd[0]; __syncthreads();
    sd[threadIdx.x] = nvp; __syncthreads();
    for (int s = 128; s > 0; s >>= 1) {
        if (threadIdx.x < s) sd[threadIdx.x] += sd[threadIdx.x + s];
        __syncthreads();
    }
    if (threadIdx.x == 0) {
        float nv = sd[0];
        float l = (nv > 0.f) ? loss / fmaxf(nv, 1.0f) : loss;
        out[0] = tw[0] * l;
    }
}

// ---------------------------------------------------------------------------
extern "C" void kernel_launch(void* const* d_in, const int* in_sizes, int n_in,
                              void* d_out, int out_size, void* d_ws, size_t ws_size,
                              hipStream_t stream) {
    (void)in_sizes; (void)n_in; (void)out_size; (void)ws_size;
    const float* preds      = (const float*)d_in[0];   // [8,21,64,64]
    const float* low_feats  = (const float*)d_in[1];   // [8,3,256,256]
    const float* high_feats = (const float*)d_in[2];   // [8,256,64,64]
    const int*   ROIs       = (const int*)d_in[3];     // [8,256,256]
    const float* tw         = (const float*)d_in[4];   // [1]
    const float* pred_pred  = (const float*)d_in[5];   // [8,21,64,64]
    float* out = (float*)d_out;

    char* ws = (char*)d_ws;
    float* Rh      = (float*)(ws);                  //  64x256
    float* RwT     = (float*)(ws + 65536);          // 256x64
    float* tmp1    = (float*)(ws + 131072);         // 24 x 64x256
    float* lowr    = (float*)(ws + 1703936);        // 24 x 64x64
    float* roi_ds  = (float*)(ws + 2097152);        // 8 x 4096
    int*   order1  = (int*)(ws + 2228224);
    int*   par1    = (int*)(ws + 2359296);
    int*   order2  = (int*)(ws + 2490368);
    int*   par2    = (int*)(ws + 2621440);
    float* wedge1  = (float*)(ws + 2752512);
    float* wedge2  = (float*)(ws + 2883584);
    float* gtmp    = (float*)(ws + 3014656);        // 8 x 4096 x 22
    float* AS      = (float*)(ws + 5898240);        // 8 x 4096 x 21
    float* partials= (float*)(ws + 8650752);        // 512

    // interpolation matrices (Rh and RwT are contiguous: zero both at once)
    tel_zero_kernel<<<128, 256, 0, stream>>>(Rh, 32768);
    tel_interp_kernel<<<1, 64, 0, stream>>>(Rh, RwT);

    // resize: tmp1 = Rh * X ; lowr = tmp1 * RwT   (24 batched f32 WMMA GEMMs)
    tel_gemm_wmma_f32<<<dim3(64, 24), 32, 0, stream>>>(
        Rh, low_feats, tmp1, 256, 256, 256, 256, 256, 0L, 65536L, 16384L);
    tel_gemm_wmma_f32<<<dim3(16, 24), 32, 0, stream>>>(
        tmp1, RwT, lowr, 64, 256, 256, 64, 64, 16384L, 0L, 4096L);

    tel_roi_kernel<<<128, 256, 0, stream>>>(ROIs, roi_ds);

    // MSTs (low-level tree, high-level tree)
    tel_build_tree<<<BATCH, 256, 180224, stream>>>(lowr, 3, 3L * NPIX, order1, par1);
    tel_build_tree<<<BATCH, 256, 180224, stream>>>(high_feats, 256, 256L * NPIX, order2, par2);

    tel_wedge_kernel<<<128, 256, 0, stream>>>(lowr, 3, 3L * NPIX, par1, 0.002f, wedge1);
    tel_wedge_kernel<<<128, 256, 0, stream>>>(high_feats, 256, 256L * NPIX, par2, 1.0f, wedge2);

    // pass 1: filter preds on low tree
    tel_tree_filter_chunk<<<BATCH, 256, 229376, stream>>>(preds, 0, 0,  order1, par1, wedge1, gtmp);
    tel_tree_filter_chunk<<<BATCH, 256, 229376, stream>>>(preds, 0, 11, order1, par1, wedge1, gtmp);
    tel_divide_kernel<<<2688, 256, 0, stream>>>(gtmp, AS);

    // pass 2: filter AS on high tree
    tel_tree_filter_chunk<<<BATCH, 256, 229376, stream>>>(AS, 1, 0,  order2, par2, wedge2, gtmp);
    tel_tree_filter_chunk<<<BATCH, 256, 229376, stream>>>(AS, 1, 11, order2, par2, wedge2, gtmp);
    tel_divide_kernel<<<2688, 256, 0, stream>>>(gtmp, AS);

    // loss
    tel_loss_partial<<<512, 256, 0, stream>>>(pred_pred, AS, roi_ds, partials);
    tel_loss_final<<<1, 256, 0, stream>>>(partials, roi_ds, tw, out);
}